// Spline3D_80470507258267
// MI455X (gfx1250) — compile-verified
//
#include <hip/hip_runtime.h>

typedef __attribute__((ext_vector_type(2))) float v2f;
typedef __attribute__((ext_vector_type(4))) float f4;
typedef __attribute__((ext_vector_type(8))) float v8f;

#define GRID 64   // knots is GRID^3 x 4 channels (fixed by reference)

// One wave (32 lanes) processes 16 points; lanes (p, p+16) form a team for
// point p.  Basis coefficients for all 16 points in a dim come from ONE
// v_wmma_f32_16x16x4_f32:
//   A-operand rows 0-3 and 8-11 = basis^T (so both half-wave D rows carry it),
//   B-operand col p            = [1, s_p, s_p^2, s_p^3],
//   => every lane reads c0..c3 of its point from D components 0..3.
__global__ __launch_bounds__(256) void
spline3d_wmma_kernel(const float* __restrict__ coords,
                     const float* __restrict__ knots,
                     float* __restrict__ out, int n_points)
{
    const int lane = threadIdx.x & 31;
    const int half = lane >> 4;          // which lane of the 2-lane team
    const int sub  = lane & 15;          // point slot within the wave
    const int gwave = (blockIdx.x * (int)blockDim.x + (int)threadIdx.x) >> 5;
    const int n_orig = gwave * 16 + sub;
    int n = n_orig < n_points ? n_orig : n_points - 1;  // keep EXEC all-ones

    // Catmull-Rom basis, transposed: Bt[j][k] = A[k][j], c_j(s) = sum_k Bt[j][k] s^k
    const float Bt[4][4] = {
        { 0.0f, -0.5f,  1.0f, -0.5f},
        { 1.0f,  0.0f, -2.5f,  1.5f},
        { 0.0f,  0.5f,  2.0f, -1.5f},
        { 0.0f,  0.0f, -0.5f,  0.5f},
    };
    // A-operand (16x4 f32, 2 VGPRs): lane l covers row M=l&15, K = {0,1} for
    // lanes 0-15 and K = {2,3} for lanes 16-31.  Rows 0-3 and 8-11 = Bt.
    const bool arow = ((sub & 7) < 4);
    const int  r    = sub & 3;
    v2f aop;
    aop.x = arow ? Bt[r][half ? 2 : 0] : 0.0f;
    aop.y = arow ? Bt[r][half ? 3 : 1] : 0.0f;

    // Both team lanes load the same point's coords (WGP-cache hit).
    const float czc = coords[3 * n + 0];
    const float cyc = coords[3 * n + 1];
    const float cxc = coords[3 * n + 2];
    const int iz = (int)czc, iy = (int)cyc, ix = (int)cxc;
    float sz = czc - (float)iz, sy = cyc - (float)iy, sx = cxc - (float)ix;
    sz = fminf(fmaxf(sz, 0.0f), 1.0f);
    sy = fminf(fmaxf(sy, 0.0f), 1.0f);
    sx = fminf(fmaxf(sx, 0.0f), 1.0f);
    int z0 = iz - 1, y0 = iy - 1, x0 = ix - 1;
    z0 = max(0, min(z0, GRID - 4));
    y0 = max(0, min(y0, GRID - 4));
    x0 = max(0, min(x0, GRID - 4));

    // B-operand (4x16 f32, 2 VGPRs): VGPR0 = rows K=0 / K=2, VGPR1 = K=1 / K=3.
    const v8f zero8 = {};
    v2f bz, by, bx;
    bz.x = half ? sz * sz : 1.0f;  bz.y = half ? sz * sz * sz : sz;
    by.x = half ? sy * sy : 1.0f;  by.y = half ? sy * sy * sy : sy;
    bx.x = half ? sx * sx : 1.0f;  bx.y = half ? sx * sx * sx : sx;

    v8f dz8 = __builtin_amdgcn_wmma_f32_16x16x4_f32(
        false, aop, false, bz, (short)0, zero8, false, false);
    v8f dy8 = __builtin_amdgcn_wmma_f32_16x16x4_f32(
        false, aop, false, by, (short)0, zero8, false, false);
    v8f dx8 = __builtin_amdgcn_wmma_f32_16x16x4_f32(
        false, aop, false, bx, (short)0, zero8, false, false);

    const float wx0 = dx8[0], wx1 = dx8[1], wx2 = dx8[2], wx3 = dx8[3];
    const float wy[4] = { dy8[0], dy8[1], dy8[2], dy8[3] };
    // Team split over z: lane p -> dz {0,1}, lane p+16 -> dz {2,3}.
    const float wzl[2] = { half ? dz8[2] : dz8[0], half ? dz8[3] : dz8[1] };
    const int   zbase  = z0 + 2 * half;

    f4 acc = {0.0f, 0.0f, 0.0f, 0.0f};
#pragma unroll
    for (int dz2 = 0; dz2 < 2; ++dz2) {
        const int z = zbase + dz2;
        const float wz = wzl[dz2];
#pragma unroll
        for (int dy = 0; dy < 4; ++dy) {
            // 4 cells x 4 channels = 64 contiguous, 16B-aligned bytes
            const f4* kp = (const f4*)(knots +
                (size_t)((((z * GRID) + (y0 + dy)) * GRID) + x0) * 4);
            f4 k0 = kp[0], k1 = kp[1], k2 = kp[2], k3 = kp[3];
            f4 t = wx0 * k0 + wx1 * k1 + wx2 * k2 + wx3 * k3;
            acc += (wz * wy[dy]) * t;
        }
    }

    // Pair reduction across the two team lanes (wave32 -> xor by 16).
    acc.x += __shfl_xor(acc.x, 16, 32);
    acc.y += __shfl_xor(acc.y, 16, 32);
    acc.z += __shfl_xor(acc.z, 16, 32);
    acc.w += __shfl_xor(acc.w, 16, 32);

    if (half == 0 && n_orig < n_points) {
        f4* op = (f4*)(out + (size_t)n * 4);
        *op = acc;   // global_store_b128, 16B aligned
    }
}

extern "C" void kernel_launch(void* const* d_in, const int* in_sizes, int n_in,
                              void* d_out, int out_size, void* d_ws, size_t ws_size,
                              hipStream_t stream) {
    const float* coords = (const float*)d_in[0];   // (N, 3) f32
    const float* knots  = (const float*)d_in[1];   // (64, 64, 64, 4) f32
    float* out = (float*)d_out;                    // (N, 4) f32

    const int n_points = in_sizes[0] / 3;
    const int waves   = (n_points + 15) / 16;      // 16 points per wave
    const int threads = waves * 32;
    const int block   = 256;                       // 8 waves / block
    const int grid    = (threads + block - 1) / block;

    spline3d_wmma_kernel<<<grid, block, 0, stream>>>(coords, knots, out, n_points);
}